// ClusterProcessor_63917703299105
// MI455X (gfx1250) — compile-verified
//
#include <hip/hip_runtime.h>
#include <hip/hip_bf16.h>

// ---------------------------------------------------------------------------
// Mamba block forward for MI455X (gfx1250), fp32 end-to-end.
// GEMMs: V_WMMA_F32_16X16X4_F32 fed from LDS, staged with
// GLOBAL_LOAD_ASYNC_TO_LDS_B128 (double-buffered, ASYNCcnt-tracked).
// ---------------------------------------------------------------------------

#define BATCH   4
#define SEQLEN  2048
#define DMODEL  512
#define DINNER  1024
#define DSTATE  16
#define DTRANK  32
#define DCONV   4
#define BL      (BATCH * SEQLEN)      // 8192 rows
#define LN_EPS  1e-5f

#define BM 128                        // rows per block (8 waves x 16)
#define BN 64                         // cols per block (4 accumulators)
#define KC 16                        // K chunk staged in LDS
#define SA (KC + 4)                   // padded LDS row stride (20 floats = 80B,
                                      // 16B-aligned rows, conflict-free b64 reads)

typedef __attribute__((ext_vector_type(2))) float v2f;
typedef __attribute__((ext_vector_type(8))) float v8f;
typedef int v4i_g __attribute__((vector_size(16)));   // GCC-style vec, matches builtin

// ---------------------------------------------------------------------------
// Async global -> LDS staging helpers (CDNA5 GLOBAL_LOAD_ASYNC_TO_LDS_B128).
// Falls back to a synchronous register copy if the builtin is unavailable.
// ---------------------------------------------------------------------------
__device__ __forceinline__ void stage_b128(const float* g, float* l)
{
#if __has_builtin(__builtin_amdgcn_global_load_async_to_lds_b128)
    __builtin_amdgcn_global_load_async_to_lds_b128(
        (__attribute__((address_space(1))) v4i_g*)g,
        (__attribute__((address_space(3))) v4i_g*)l,
        0 /*offset*/, 0 /*cpol*/);
#else
    *(float4*)l = *(const float4*)g;
#endif
}

__device__ __forceinline__ void stage_wait()
{
#if __has_builtin(__builtin_amdgcn_s_wait_asynccnt)
    __builtin_amdgcn_s_wait_asynccnt(0);
#elif __has_builtin(__builtin_amdgcn_global_load_async_to_lds_b128)
    asm volatile("s_wait_asynccnt 0x0" ::: "memory");
#endif
}

// ---------------------------------------------------------------------------
// fp32 WMMA GEMM:  C[M,N] = A[M,K] * W[N,K]^T   (row-major, strided)
// Block: 256 threads (8 waves). Tile: BM x BN, K in KC chunks, double-buffered
// LDS staging via async loads. Wave w computes rows [w*16, w*16+16) x BN cols.
//
// Fragment layouts per CDNA5 ISA 7.12.2:
//   A 16x4 f32: lane(0..15)=M, half=lane>>4 selects K pair:
//       vgpr0 = A[M, k+2*half], vgpr1 = A[M, k+2*half+1]
//   B 4x16 f32 (KxN): lane(0..15)=N, half selects K pair (mirror of A).
//   C/D 16x16 f32: vgpr i, lane L -> M = i + 8*(L>>4), N = L&15.
// ---------------------------------------------------------------------------
__global__ __launch_bounds__(256)
void gemm_wmma_f32(const float* __restrict__ A,
                   const float* __restrict__ W,
                   float* __restrict__ C,
                   int K, int lda, int ldw, int ldc)
{
    __shared__ float As[2][BM * SA];   // 2 x 10 KB
    __shared__ float Ws[2][BN * SA];   // 2 x  5 KB

    const int mtile = blockIdx.x * BM;
    const int ntile = blockIdx.y * BN;
    const int tid   = threadIdx.x;     // 0..255
    const int wave  = tid >> 5;        // 0..7
    const int lane  = tid & 31;
    const int half  = lane >> 4;       // 0/1
    const int lrow  = lane & 15;

    // staging assignments: A tile = 128 rows x 16 floats (2 threads/row, 8 floats each)
    //                      W tile =  64 rows x 16 floats (4 threads/row, 4 floats each)
    const int ar = tid >> 1;           // 0..127
    const int ak = (tid & 1) * 8;      // 0 or 8
    const int wr = tid >> 2;           // 0..63
    const int wk = (tid & 3) * 4;      // 0,4,8,12

    const float* Ag = A + (size_t)(mtile + ar) * lda + ak;
    const float* Wg = W + (size_t)(ntile + wr) * ldw + wk;
    float* Al = &As[0][0] + ar * SA + ak;
    float* Wl = &Ws[0][0] + wr * SA + wk;
    const int lbuf = BM * SA;          // As buffer stride (floats)
    const int wbuf = BN * SA;

    v8f acc[4];
    #pragma unroll
    for (int j = 0; j < 4; ++j)
        acc[j] = (v8f){0.f,0.f,0.f,0.f,0.f,0.f,0.f,0.f};

    // prologue: stage chunk 0 into buffer 0
    stage_b128(Ag,     Al);
    stage_b128(Ag + 4, Al + 4);
    stage_b128(Wg,     Wl);
    stage_wait();
    __syncthreads();

    const int NC = K / KC;
    for (int kc = 0; kc < NC; ++kc) {
        const int cur = kc & 1;

        // stream next chunk into the other buffer while computing this one
        if (kc + 1 < NC) {
            const int nxt = cur ^ 1;
            const int k0  = (kc + 1) * KC;
            stage_b128(Ag + k0,     Al + nxt * lbuf);
            stage_b128(Ag + k0 + 4, Al + nxt * lbuf + 4);
            stage_b128(Wg + k0,     Wl + nxt * wbuf);
        }

        // compute from LDS buffer `cur`
        const float* Ab = &As[cur][(wave * 16 + lrow) * SA + 2 * half];
        const float* Bb = &Ws[cur][lrow * SA + 2 * half];
        #pragma unroll
        for (int kk = 0; kk < KC / 4; ++kk) {
            v2f a = *(const v2f*)(Ab + kk * 4);
            #pragma unroll
            for (int j = 0; j < 4; ++j) {
                v2f b = *(const v2f*)(Bb + j * 16 * SA + kk * 4);
                acc[j] = __builtin_amdgcn_wmma_f32_16x16x4_f32(
                    false, a, false, b, (short)0, acc[j], false, false);
            }
        }

        stage_wait();
        __syncthreads();
    }

    // store D: acc[j][i] -> row (mtile + wave*16 + i + 8*half), col (ntile + j*16 + lrow)
    #pragma unroll
    for (int i = 0; i < 8; ++i) {
        const size_t r = (size_t)(mtile + wave * 16 + i + 8 * half) * ldc;
        #pragma unroll
        for (int j = 0; j < 4; ++j)
            C[r + ntile + j * 16 + lrow] = acc[j][i];
    }
}

// ---------------------------------------------------------------------------
// Causal depthwise conv1d (k=4) + bias + SiLU.
// Reads the x half of xz (row stride 2*DINNER), writes xc (row stride DINNER).
// ---------------------------------------------------------------------------
__global__ void conv_silu_kernel(const float* __restrict__ xz,
                                 const float* __restrict__ cw,
                                 const float* __restrict__ cb,
                                 float* __restrict__ xc)
{
    const int idx = blockIdx.x * blockDim.x + threadIdx.x;
    if (idx >= BL * DINNER) return;
    const int d = idx & (DINNER - 1);
    const int m = idx >> 10;               // row in [0, BL)
    const int b = m / SEQLEN;
    const int l = m - b * SEQLEN;

    float acc = cb[d];
    #pragma unroll
    for (int j = 0; j < DCONV; ++j) {
        const int ll = l - (DCONV - 1) + j;
        if (ll >= 0)
            acc = fmaf(cw[d * DCONV + j],
                       xz[(size_t)(b * SEQLEN + ll) * (2 * DINNER) + d], acc);
    }
    const float s = acc / (1.f + expf(-acc));   // silu
    xc[(size_t)m * DINNER + d] = s;
}

// ---------------------------------------------------------------------------
// dt epilogue: dt = softplus(dt_raw + bias[d])
// ---------------------------------------------------------------------------
__global__ void dt_softplus_kernel(float* __restrict__ dt,
                                   const float* __restrict__ bias)
{
    const int i = blockIdx.x * blockDim.x + threadIdx.x;
    if (i >= BL * DINNER) return;
    const int d = i & (DINNER - 1);
    const float x = dt[i] + bias[d];
    dt[i] = (x > 20.f) ? x : log1pf(expf(x));
}

// ---------------------------------------------------------------------------
// Selective scan. One thread per (b, d): 16-wide state in registers,
// sequential over L. Fused with D-skip and y * silu(z) gating.
// Writes result in place over xc.
// ---------------------------------------------------------------------------
__global__ void scan_kernel(const float* __restrict__ dt,
                            const float* __restrict__ xdbl,
                            const float* __restrict__ xz,
                            const float* __restrict__ A_log,
                            const float* __restrict__ D_skip,
                            float* __restrict__ xc)
{
    const int g = blockIdx.x * blockDim.x + threadIdx.x;
    if (g >= BATCH * DINNER) return;
    const int d = g & (DINNER - 1);
    const int b = g >> 10;

    float An[DSTATE];
    #pragma unroll
    for (int n = 0; n < DSTATE; ++n)
        An[n] = -expf(A_log[d * DSTATE + n]);
    const float Dk = D_skip[d];

    float h[DSTATE];
    #pragma unroll
    for (int n = 0; n < DSTATE; ++n) h[n] = 0.f;

    for (int l = 0; l < SEQLEN; ++l) {
        const size_t m = (size_t)b * SEQLEN + l;
        const float dtv = dt[m * DINNER + d];
        const float u   = xc[m * DINNER + d];
        const float du  = dtv * u;
        float y = 0.f;
        #pragma unroll
        for (int n = 0; n < DSTATE; ++n) {
            const float Bn = xdbl[m * 64 + DTRANK + n];
            const float Cn = xdbl[m * 64 + DTRANK + DSTATE + n];
            const float dA = expf(dtv * An[n]);
            h[n] = fmaf(dA, h[n], du * Bn);
            y = fmaf(h[n], Cn, y);
        }
        const float z  = xz[m * (2 * DINNER) + DINNER + d];
        const float yv = fmaf(u, Dk, y);
        xc[m * DINNER + d] = yv * (z / (1.f + expf(-z)));  // y * silu(z)
    }
}

// ---------------------------------------------------------------------------
// In-place LayerNorm over DMODEL=512. One wave per row; 16 elems/lane;
// wave32 shfl_xor reductions.
// ---------------------------------------------------------------------------
__global__ void layernorm_kernel(float* __restrict__ out,
                                 const float* __restrict__ w,
                                 const float* __restrict__ bia)
{
    const int row  = blockIdx.x;
    const int lane = threadIdx.x;          // 0..31
    float v[16];
    float s = 0.f;
    #pragma unroll
    for (int i = 0; i < 16; ++i) {
        v[i] = out[(size_t)row * DMODEL + lane + i * 32];
        s += v[i];
    }
    #pragma unroll
    for (int off = 16; off > 0; off >>= 1) s += __shfl_xor(s, off, 32);
    const float mu = s * (1.f / DMODEL);

    float var = 0.f;
    #pragma unroll
    for (int i = 0; i < 16; ++i) {
        const float dd = v[i] - mu;
        var = fmaf(dd, dd, var);
    }
    #pragma unroll
    for (int off = 16; off > 0; off >>= 1) var += __shfl_xor(var, off, 32);
    const float inv = rsqrtf(var * (1.f / DMODEL) + LN_EPS);

    #pragma unroll
    for (int i = 0; i < 16; ++i) {
        const int c = lane + i * 32;
        out[(size_t)row * DMODEL + c] = (v[i] - mu) * inv * w[c] + bia[c];
    }
}

// ---------------------------------------------------------------------------
// Host-side launch
// ---------------------------------------------------------------------------
extern "C" void kernel_launch(void* const* d_in, const int* in_sizes, int n_in,
                              void* d_out, int out_size, void* d_ws, size_t ws_size,
                              hipStream_t stream)
{
    const float* x_in      = (const float*)d_in[0];   // (4,2048,512)
    const float* in_proj_w = (const float*)d_in[1];   // (2048,512)
    const float* conv_w    = (const float*)d_in[2];   // (1024,1,4)
    const float* conv_b    = (const float*)d_in[3];   // (1024,)
    const float* x_proj_w  = (const float*)d_in[4];   // (64,1024)
    const float* dt_proj_w = (const float*)d_in[5];   // (1024,32)
    const float* dt_proj_b = (const float*)d_in[6];   // (1024,)
    const float* A_log     = (const float*)d_in[7];   // (1024,16)
    const float* D_skip    = (const float*)d_in[8];   // (1024,)
    const float* out_proj_w= (const float*)d_in[9];   // (512,1024)
    const float* ln_w      = (const float*)d_in[10];  // (512,)
    const float* ln_b      = (const float*)d_in[11];  // (512,)
    float* out = (float*)d_out;                       // (4,2048,512)

    // workspace layout (floats)
    float* ws   = (float*)d_ws;
    float* xz   = ws;                                  // BL * 2048
    float* xc   = xz  + (size_t)BL * 2 * DINNER;       // BL * 1024  (x, later gated y)
    float* xdbl = xc  + (size_t)BL * DINNER;           // BL * 64
    float* dt   = xdbl+ (size_t)BL * 64;               // BL * 1024

    // 1) in_proj: xz = x_in @ in_proj_w^T    M=8192 N=2048 K=512
    gemm_wmma_f32<<<dim3(BL / BM, (2 * DINNER) / BN), 256, 0, stream>>>(
        x_in, in_proj_w, xz, DMODEL, DMODEL, DMODEL, 2 * DINNER);

    // 2) causal depthwise conv + SiLU -> xc
    conv_silu_kernel<<<(BL * DINNER + 255) / 256, 256, 0, stream>>>(
        xz, conv_w, conv_b, xc);

    // 3) x_proj: xdbl = xc @ x_proj_w^T      M=8192 N=64 K=1024
    gemm_wmma_f32<<<dim3(BL / BM, 64 / BN), 256, 0, stream>>>(
        xc, x_proj_w, xdbl, DINNER, DINNER, DINNER, 64);

    // 4) dt_proj: dt = xdbl[:, :32] @ dt_proj_w^T   M=8192 N=1024 K=32 (lda=64)
    gemm_wmma_f32<<<dim3(BL / BM, DINNER / BN), 256, 0, stream>>>(
        xdbl, dt_proj_w, dt, DTRANK, 64, DTRANK, DINNER);

    // 5) dt = softplus(dt + bias)
    dt_softplus_kernel<<<(BL * DINNER + 255) / 256, 256, 0, stream>>>(dt, dt_proj_b);

    // 6) selective scan + D-skip + silu(z) gating, in place over xc
    scan_kernel<<<(BATCH * DINNER) / 128, 128, 0, stream>>>(
        dt, xdbl, xz, A_log, D_skip, xc);

    // 7) out_proj: out = xc @ out_proj_w^T   M=8192 N=512 K=1024
    gemm_wmma_f32<<<dim3(BL / BM, DMODEL / BN), 256, 0, stream>>>(
        xc, out_proj_w, out, DINNER, DINNER, DINNER, DMODEL);

    // 8) LayerNorm in place on out
    layernorm_kernel<<<BL, 32, 0, stream>>>(out, ln_w, ln_b);
}